// GraphModel_40441412059561
// MI455X (gfx1250) — compile-verified
//
#include <hip/hip_runtime.h>

// GCNConv pipeline for gfx1250:
//   1) deg init (self loop) + node mask init
//   2) mark referenced nodes from x
//   3) degree count (f32 atomics)
//   4) deg -> rsqrt in place
//   5) xw = features @ W   via V_WMMA_F32_16X16X4_F32 (W swizzled in LDS, b64 frags)
//   6) nout = xw * dinv^2 + b   (self-loop term + bias)
//   7) edge scatter: nout[dst] += xw[src] * dinv[src]*dinv[dst]  (masked, f32 atomics, L2-resident)
//   8) out = nout[x]

typedef __attribute__((ext_vector_type(2))) float v2f;
typedef __attribute__((ext_vector_type(8))) float v8f;

#define F_IN 256
#define D_OUT 64

__global__ void k_init(float* __restrict__ deg, unsigned char* __restrict__ mask, int n) {
  int i = blockIdx.x * blockDim.x + threadIdx.x;
  if (i < n) { deg[i] = 1.0f; mask[i] = 0; }
}

__global__ void k_mark(const int* __restrict__ x, unsigned char* __restrict__ mask, int nx) {
  int i = blockIdx.x * blockDim.x + threadIdx.x;
  if (i < nx) mask[x[i]] = 1;
}

__global__ void k_degcount(const int* __restrict__ dst, float* __restrict__ deg, int ne) {
  int i = blockIdx.x * blockDim.x + threadIdx.x;
  if (i < ne) atomicAdd(&deg[dst[i]], 1.0f);
}

__global__ void k_rsqrt(float* __restrict__ deg, int n) {
  int i = blockIdx.x * blockDim.x + threadIdx.x;
  if (i < n) deg[i] = rsqrtf(deg[i]);
}

// xw[N,64] = features[N,256] @ W[256,64], fp32 WMMA 16x16x4.
// Block: 256 threads = 8 waves; wave w owns rows [blk*128 + 16w, +16).
// W is staged in LDS K-pair-swizzled: Wl[((k>>1)*64 + n)*2 + (k&1)] = W[k][n],
// so every WMMA B fragment is one aligned ds_load_b64 (no register shuffles).
__global__ __launch_bounds__(256) void k_gemm(const float* __restrict__ F,
                                              const float* __restrict__ Wg,
                                              float* __restrict__ XW, int n) {
  __shared__ float Wl[F_IN * D_OUT];  // 64 KB, swizzled
  int t = threadIdx.x;
  for (int idx = t; idx < (F_IN / 2) * D_OUT; idx += 256) {
    int p = idx / D_OUT;          // k-pair index
    int c = idx % D_OUT;          // column
    float2 pr;
    pr.x = Wg[(2 * p)     * D_OUT + c];
    pr.y = Wg[(2 * p + 1) * D_OUT + c];
    *(float2*)(Wl + (size_t)idx * 2) = pr;
  }
  __syncthreads();

  int wave = t >> 5;        // wave32
  int lane = t & 31;
  int m  = lane & 15;       // row-in-tile for A, col-in-tile for B, N for C/D
  int kh = lane >> 4;       // which K-pair of the K=4 slab this lane holds
  int r0 = blockIdx.x * 128 + wave * 16;
  int row  = r0 + m;
  int rowc = row < n ? row : (n - 1);   // clamp loads; stores are guarded
  const float* frow = F + (size_t)rowc * F_IN;

  v8f acc0 = {}, acc1 = {}, acc2 = {}, acc3 = {};

  for (int k0 = 0; k0 < F_IN; k0 += 4) {
    int ka = k0 + 2 * kh;                          // this lane holds K = ka, ka+1
    v2f a = *(const v2f*)(frow + ka);              // A(m,ka), A(m,ka+1): one b64

    // B(k,n): lane = n + 16*(k>>1 mod 2), reg = k&1 -> contiguous pair in Wl
    const v2f* wp = (const v2f*)Wl + ((k0 >> 1) + kh) * D_OUT + m;
    v2f b0 = wp[0];                                // cols  0..15
    v2f b1 = wp[16];                               // cols 16..31
    v2f b2 = wp[32];                               // cols 32..47
    v2f b3 = wp[48];                               // cols 48..63

    acc0 = __builtin_amdgcn_wmma_f32_16x16x4_f32(false, a, false, b0, (short)0, acc0, false, false);
    acc1 = __builtin_amdgcn_wmma_f32_16x16x4_f32(false, a, false, b1, (short)0, acc1, false, false);
    acc2 = __builtin_amdgcn_wmma_f32_16x16x4_f32(false, a, false, b2, (short)0, acc2, false, false);
    acc3 = __builtin_amdgcn_wmma_f32_16x16x4_f32(false, a, false, b3, (short)0, acc3, false, false);
  }

  // C/D layout: VGPR r holds (M = r + 8*(lane>>4), N = lane&15)
  for (int r = 0; r < 8; ++r) {
    int orow = r0 + r + 8 * kh;
    if (orow < n) {
      float* o = XW + (size_t)orow * D_OUT + m;
      o[0]  = acc0[r];
      o[16] = acc1[r];
      o[32] = acc2[r];
      o[48] = acc3[r];
    }
  }
}

// nout[i] = xw[i] * dinv[i]^2 + b   (self-loop message + bias)
__global__ void k_initout(const float* __restrict__ xw, const float* __restrict__ dinv,
                          const float* __restrict__ b, float* __restrict__ nout, int n) {
  long long tid = (long long)blockIdx.x * blockDim.x + threadIdx.x;
  int i = (int)(tid >> 4);
  int q = (int)(tid & 15);
  if (i >= n) return;
  float di = dinv[i];
  float w = di * di;
  float4 v  = ((const float4*)(xw + (size_t)i * D_OUT))[q];
  float4 bb = ((const float4*)b)[q];
  float4 r;
  r.x = v.x * w + bb.x; r.y = v.y * w + bb.y;
  r.z = v.z * w + bb.z; r.w = v.w * w + bb.w;
  ((float4*)(nout + (size_t)i * D_OUT))[q] = r;
}

// 16 lanes per edge, 4 dims per lane; skip edges whose dst is never read.
__global__ void k_scatter(const int* __restrict__ src, const int* __restrict__ dst,
                          const float* __restrict__ dinv, const float* __restrict__ xw,
                          const unsigned char* __restrict__ mask,
                          float* __restrict__ nout, int ne) {
  long long tid = (long long)blockIdx.x * blockDim.x + threadIdx.x;
  int e = (int)(tid >> 4);
  int q = (int)(tid & 15);
  if (e >= ne) return;
  int d = dst[e];
  if (!mask[d]) return;
  int s = src[e];
  float w = dinv[s] * dinv[d];
  float4 v = ((const float4*)(xw + (size_t)s * D_OUT))[q];
  float* o = nout + (size_t)d * D_OUT + 4 * q;
  atomicAdd(o + 0, v.x * w);
  atomicAdd(o + 1, v.y * w);
  atomicAdd(o + 2, v.z * w);
  atomicAdd(o + 3, v.w * w);
}

__global__ void k_gather(const int* __restrict__ x, const float* __restrict__ nout,
                         float* __restrict__ out, int nx) {
  long long tid = (long long)blockIdx.x * blockDim.x + threadIdx.x;
  int r = (int)(tid >> 4);
  int q = (int)(tid & 15);
  if (r >= nx) return;
  int node = x[r];
  ((float4*)(out + (size_t)r * D_OUT))[q] =
      ((const float4*)(nout + (size_t)node * D_OUT))[q];
}

extern "C" void kernel_launch(void* const* d_in, const int* in_sizes, int n_in,
                              void* d_out, int out_size, void* d_ws, size_t ws_size,
                              hipStream_t stream) {
  const float* features = (const float*)d_in[0];
  const int*   edges    = (const int*)d_in[1];
  const float* W        = (const float*)d_in[2];
  const float* b        = (const float*)d_in[3];
  const int*   x        = (const int*)d_in[4];

  int n  = in_sizes[0] / F_IN;   // 100000 nodes
  int ne = in_sizes[1] / 2;      // 3.2M edges
  int nx = in_sizes[4];          // 81920 gathered rows

  const int* src = edges;
  const int* dst = edges + ne;

  // Workspace layout: deg/dinv [n f32] | mask [n u8] | xw [n*64 f32] | nout [n*64 f32]
  char* ws = (char*)d_ws;
  float* deg = (float*)ws;
  unsigned char* mask = (unsigned char*)(ws + (size_t)n * 4);
  size_t off = ((size_t)n * 5 + 255) & ~(size_t)255;
  float* xw   = (float*)(ws + off);
  float* nout = (float*)(ws + off + (size_t)n * D_OUT * 4);

  int B = 256;
  k_init    <<<(n + B - 1) / B, B, 0, stream>>>(deg, mask, n);
  k_mark    <<<(nx + B - 1) / B, B, 0, stream>>>(x, mask, nx);
  k_degcount<<<(ne + B - 1) / B, B, 0, stream>>>(dst, deg, ne);
  k_rsqrt   <<<(n + B - 1) / B, B, 0, stream>>>(deg, n);
  k_gemm    <<<(n + 127) / 128, 256, 0, stream>>>(features, W, xw, n);
  k_initout <<<(int)(((long long)n * 16 + B - 1) / B), B, 0, stream>>>(xw, deg, b, nout, n);
  k_scatter <<<(int)(((long long)ne * 16 + B - 1) / B), B, 0, stream>>>(src, dst, deg, xw, mask, nout, ne);
  k_gather  <<<(int)(((long long)nx * 16 + B - 1) / B), B, 0, stream>>>(x, nout, (float*)d_out, nx);
}